// SSIM3D_13546326851854
// MI455X (gfx1250) — compile-verified
//
#include <hip/hip_runtime.h>

// ---------------------------------------------------------------------------
// 3D SSIM on MI455X (gfx1250), compile-only target.
// Separable 11-tap Gaussian conv expressed as banded fp32 WMMA GEMMs:
//   out(16 outs x 16 lines) = band(16x28) x in(28x16)  via 7x V_WMMA_F32_16X16X4_F32
// 3 passes (D, W, H) + fused SSIM/reduction in pass 3. Memory-bound problem,
// so full fp32 matrix path (precision is free vs 23.3 TB/s HBM ceiling).
// ---------------------------------------------------------------------------

typedef float v2f __attribute__((ext_vector_type(2)));
typedef float v8f __attribute__((ext_vector_type(8)));

#define NS     160
#define NBAT   2
#define NVOX   ((size_t)NBAT * NS * NS * NS)   /* 8,192,000 per channel */
#define TILES  32000                           /* 2 * 160 * 10 * 10     */
#define WPB    4                               /* waves (tiles) / block */
#define BLOCKS (TILES / WPB)

// Normalized 1D Gaussian, KS=11, sigma=1.5 (product of three == normalized 3D kernel)
__device__ __constant__ float W1D[11] = {
    0.00102838f, 0.00759880f, 0.03600078f, 0.10936070f, 0.21300553f,
    0.26601170f,
    0.21300553f, 0.10936070f, 0.03600078f, 0.00759880f, 0.00102838f};

__device__ __forceinline__ int refl(int x) {          // jnp.pad 'reflect'
    x = (x < 0) ? -x : x;
    return (x > NS - 1) ? (2 * (NS - 1) - x) : x;
}

// Band matrix element: Wband[m][k] = W1D[k-m] if 0 <= k-m <= 10 else 0
__device__ __forceinline__ float bandw(int m, int k) {
    int d   = k - m;
    int idx = d < 0 ? 0 : (d > 10 ? 10 : d);
    float w = W1D[idx];
    return (d >= 0 && d <= 10) ? w : 0.0f;
}

// Per-lane operand addressing shared by A(16x4) and B(4x16) fp32 WMMA layouts:
//   ln = lane & 15 (row of A / col of B), k0 = 4*q + 2*(lane>=16)
// lds tile layout: L[ch][k(0..31)][line(0..15)], pitch 17 -> no LDS bank conflicts.

// ---------------------------------------------------------------------------
// Pass 1: conv along contiguous D. Computes the 5 channels on the fly.
//   D' = A(data: 16 lines x K) x B(band^T: K x 16 outs)  => D'[line][convpos]
//   (N = conv position lands on contiguous D axis -> coalesced stores)
// ---------------------------------------------------------------------------
__global__ __launch_bounds__(128) void ssim_pass1(const float* __restrict__ src,
                                                  const float* __restrict__ ref,
                                                  float* __restrict__ out) {
    __shared__ float lds[WPB][5][32][17];
    const int lane = threadIdx.x & 31;
    const int wave = threadIdx.x >> 5;
    const int tid  = blockIdx.x * WPB + wave;
    const int dt = tid % 10;
    const int wt = (tid / 10) % 10;
    const int h  = (tid / 100) % NS;
    const int b  = tid / 16000;
    const int d0 = dt * 16, w0 = wt * 16;

    float(*L)[32][17]   = lds[wave];
    const size_t rowb   = ((size_t)b * NS + h) * NS;
    const int din       = refl(d0 - 5 + lane);       // j = lane (0..31)

    for (int n = 0; n < 16; ++n) {                   // line = w0+n
        size_t addr = (rowb + (size_t)(w0 + n)) * NS + din;
        float s = src[addr];
        float r = ref[addr];
        L[0][lane][n] = s;
        L[1][lane][n] = r;
        L[2][lane][n] = s * s;
        L[3][lane][n] = r * r;
        L[4][lane][n] = s * r;
    }

    const int ln = lane & 15;
    const int kb = (lane >> 4) << 1;                 // 0 or 2
    v2f wv[7];
#pragma unroll
    for (int q = 0; q < 7; ++q) {                    // B[k][n] = Wband[n][k]
        int k0 = 4 * q + kb;
        wv[q].x = bandw(ln, k0);
        wv[q].y = bandw(ln, k0 + 1);
    }
    const int lineadd = (lane < 16) ? 0 : 8;

    for (int ch = 0; ch < 5; ++ch) {
        v8f c = {0.f, 0.f, 0.f, 0.f, 0.f, 0.f, 0.f, 0.f};
#pragma unroll
        for (int q = 0; q < 7; ++q) {
            int k0 = 4 * q + kb;
            v2f a;                                   // A[row=ln][k] = in[k][line=ln]
            a.x = L[ch][k0][ln];
            a.y = L[ch][k0 + 1][ln];
            c = __builtin_amdgcn_wmma_f32_16x16x4_f32(false, a, false, wv[q],
                                                      (short)0, c, false, false);
        }
        float* outc = out + (size_t)ch * NVOX;
#pragma unroll
        for (int v = 0; v < 8; ++v) {                // M=line, N=conv pos (contig D)
            int line = v + lineadd;
            outc[(rowb + (size_t)(w0 + line)) * NS + (size_t)(d0 + ln)] = c[v];
        }
    }
}

// ---------------------------------------------------------------------------
// Pass 2: conv along W (stride 160), lines along contiguous D.
//   D = A(band: 16 outs x K) x B(data: K x 16 lines)  => D[convpos][line]
// ---------------------------------------------------------------------------
__global__ __launch_bounds__(128) void ssim_pass2(const float* __restrict__ in,
                                                  float* __restrict__ out) {
    __shared__ float lds[WPB][5][32][17];
    const int lane = threadIdx.x & 31;
    const int wave = threadIdx.x >> 5;
    const int tid  = blockIdx.x * WPB + wave;
    const int dt = tid % 10;
    const int wt = (tid / 10) % 10;
    const int h  = (tid / 100) % NS;
    const int b  = tid / 16000;
    const int d0 = dt * 16, w0 = wt * 16;

    float(*L)[32][17] = lds[wave];
    const size_t rowb = ((size_t)b * NS + h) * NS;

    const int n  = lane & 15;                        // line (contig D)
    const int jh = lane >> 4;                        // 0/1: two conv rows per iter
    for (int it = 0; it < 16; ++it) {
        int jj  = 2 * it + jh;
        int win = refl(w0 - 5 + jj);
        size_t addr = (rowb + (size_t)win) * NS + (size_t)(d0 + n);
        for (int ch = 0; ch < 5; ++ch)
            L[ch][jj][n] = in[(size_t)ch * NVOX + addr];
    }

    const int ln = lane & 15;
    const int kb = (lane >> 4) << 1;
    v2f wv[7];
#pragma unroll
    for (int q = 0; q < 7; ++q) {                    // A[row=ln][k] = Wband[ln][k]
        int k0 = 4 * q + kb;
        wv[q].x = bandw(ln, k0);
        wv[q].y = bandw(ln, k0 + 1);
    }
    const int madd = (lane < 16) ? 0 : 8;

    for (int ch = 0; ch < 5; ++ch) {
        v8f c = {0.f, 0.f, 0.f, 0.f, 0.f, 0.f, 0.f, 0.f};
#pragma unroll
        for (int q = 0; q < 7; ++q) {
            int k0 = 4 * q + kb;
            v2f bd;                                  // B[k][col=ln] = in[k][line=ln]
            bd.x = L[ch][k0][ln];
            bd.y = L[ch][k0 + 1][ln];
            c = __builtin_amdgcn_wmma_f32_16x16x4_f32(false, wv[q], false, bd,
                                                      (short)0, c, false, false);
        }
        float* outc = out + (size_t)ch * NVOX;
#pragma unroll
        for (int v = 0; v < 8; ++v) {                // M=conv pos (W), N=line (D)
            int mo = v + madd;
            outc[(rowb + (size_t)(w0 + mo)) * NS + (size_t)(d0 + ln)] = c[v];
        }
    }
}

// ---------------------------------------------------------------------------
// Pass 3: conv along H (stride 160*160) + fused SSIM + deterministic reduce.
// ---------------------------------------------------------------------------
__global__ __launch_bounds__(128) void ssim_pass3(const float* __restrict__ in,
                                                  float* __restrict__ partials) {
    __shared__ float lds[WPB][5][32][17];
    const int lane = threadIdx.x & 31;
    const int wave = threadIdx.x >> 5;
    const int tid  = blockIdx.x * WPB + wave;
    const int dt = tid % 10;
    const int w  = (tid / 10) % NS;
    const int ht = (tid / 1600) % 10;
    const int b  = tid / 16000;
    const int d0 = dt * 16, h0 = ht * 16;

    float(*L)[32][17] = lds[wave];

    const int n  = lane & 15;
    const int jh = lane >> 4;
    for (int it = 0; it < 16; ++it) {
        int jj  = 2 * it + jh;
        int hin = refl(h0 - 5 + jj);
        size_t addr = (((size_t)b * NS + hin) * NS + w) * NS + (size_t)(d0 + n);
        for (int ch = 0; ch < 5; ++ch)
            L[ch][jj][n] = in[(size_t)ch * NVOX + addr];
    }

    const int ln = lane & 15;
    const int kb = (lane >> 4) << 1;
    v2f wv[7];
#pragma unroll
    for (int q = 0; q < 7; ++q) {
        int k0 = 4 * q + kb;
        wv[q].x = bandw(ln, k0);
        wv[q].y = bandw(ln, k0 + 1);
    }

    v8f acc[5];
#pragma unroll
    for (int ch = 0; ch < 5; ++ch) {
        v8f c = {0.f, 0.f, 0.f, 0.f, 0.f, 0.f, 0.f, 0.f};
#pragma unroll
        for (int q = 0; q < 7; ++q) {
            int k0 = 4 * q + kb;
            v2f bd;
            bd.x = L[ch][k0][ln];
            bd.y = L[ch][k0 + 1][ln];
            c = __builtin_amdgcn_wmma_f32_16x16x4_f32(false, wv[q], false, bd,
                                                      (short)0, c, false, false);
        }
        acc[ch] = c;
    }

    // Fused SSIM on matching (VGPR, lane) slots (identical layout for all 5 ch)
    float ssum = 0.f;
#pragma unroll
    for (int v = 0; v < 8; ++v) {
        float mu1 = acc[0][v], mu2 = acc[1][v];
        float m11 = acc[2][v], m22 = acc[3][v], m12 = acc[4][v];
        float mu1sq = mu1 * mu1, mu2sq = mu2 * mu2, mu12 = mu1 * mu2;
        float s1 = m11 - mu1sq, s2 = m22 - mu2sq, s12 = m12 - mu12;
        float num = (2.f * mu12 + 1e-4f) * (2.f * s12 + 9e-4f);
        float den = (mu1sq + mu2sq + 1e-4f) * (s1 + s2 + 9e-4f);
        ssum += num / (den + 1e-12f);
    }
#pragma unroll
    for (int off = 16; off > 0; off >>= 1)
        ssum += __shfl_xor(ssum, off, 32);
    if (lane == 0) partials[tid] = ssum;             // one slot per tile: deterministic
}

__global__ __launch_bounds__(256) void ssim_finalize(const float* __restrict__ partials,
                                                     float* __restrict__ out) {
    __shared__ float sm[256];
    float s = 0.f;
    for (int i = threadIdx.x; i < TILES; i += 256) s += partials[i];
    sm[threadIdx.x] = s;
    __syncthreads();
    for (int off = 128; off > 0; off >>= 1) {
        if ((int)threadIdx.x < off) sm[threadIdx.x] += sm[threadIdx.x + off];
        __syncthreads();
    }
    if (threadIdx.x == 0) out[0] = 1.0f - sm[0] / 8192000.0f;
}

// ---------------------------------------------------------------------------
extern "C" void kernel_launch(void* const* d_in, const int* in_sizes, int n_in,
                              void* d_out, int out_size, void* d_ws, size_t ws_size,
                              hipStream_t stream) {
    (void)in_sizes; (void)n_in; (void)out_size;
    const float* src = (const float*)d_in[0];   // 'source'
    const float* ref = (const float*)d_in[1];   // 'reference'

    float* tmpA     = (float*)d_ws;             // 5ch, conv along D
    float* tmpB     = tmpA + 5 * NVOX;          // 5ch, conv along D,W
    float* partials = tmpB + 5 * NVOX;          // TILES floats

    // Need ~328 MB workspace; bail (deterministically) if not provided.
    if (ws_size < (10 * NVOX + TILES) * sizeof(float)) return;

    ssim_pass1<<<BLOCKS, 128, 0, stream>>>(src, ref, tmpA);
    ssim_pass2<<<BLOCKS, 128, 0, stream>>>(tmpA, tmpB);
    ssim_pass3<<<BLOCKS, 128, 0, stream>>>(tmpB, partials);
    ssim_finalize<<<1, 256, 0, stream>>>(partials, (float*)d_out);
}